// LossFunction_29145648071076
// MI455X (gfx1250) — compile-verified
//
#include <hip/hip_runtime.h>
#include <hip/hip_bf16.h>
#include <math.h>

// Problem constants from the reference
#define BB   128
#define NTT  64
#define NRR  64
#define KK   16
#define LLn  16
#define MMn  8
#define IIn  8
#define GG   181   // angle grid 0..180 step 1

typedef __attribute__((ext_vector_type(2))) float v2f;
typedef __attribute__((ext_vector_type(8))) float v8f;
typedef __attribute__((ext_vector_type(4))) unsigned int u32x4;
typedef __attribute__((ext_vector_type(4))) int i32x4;
typedef __attribute__((ext_vector_type(8))) int i32x8;

__global__ void zero_out_kernel(float* o) {
    if (threadIdx.x == 0 && blockIdx.x == 0) o[0] = 0.0f;
}

// One workgroup (256 threads = 8 wave32) per batch element.
__global__ __launch_bounds__(256) void loss_kernel(
    const float* __restrict__ uuPow,   // (B,K) f32
    const float* __restrict__ duCom,   // (B,L,NT) c64 -> interleaved f32
    const float* __restrict__ sens,    // (B,M,NT) c64
    const float* __restrict__ uu,      // (B,K,NR) c64
    const float* __restrict__ du,      // (B,L,NT) c64
    const float* __restrict__ ta,      // (B,M,2) c64  [ang, gain] in .re
    const float* __restrict__ inm,     // (B,I,2) c64
    const float* __restrict__ ci,      // (B,K,L) c64
    const float* __restrict__ n2du,    // (B,L,1) f32
    const float* __restrict__ n2bs,    // scalar f32
    const float* __restrict__ aSI,     // scalar f32
    const float* __restrict__ rp,      // (NR,NT) f32
    float* __restrict__ out)
{
    const int b   = blockIdx.x;
    const int tid = threadIdx.x;

    __shared__ float sX[2 * NTT];          // x = sum of beams (complex)
    __shared__ float sY[2 * NRR];          // y = Bm x (complex)
    __shared__ float sPhi[16], sGain[16], sAng[16];
    __shared__ float sSr[16], sSi[16];     // s_l = a_t(l)^H x
    __shared__ float sWr[32 * 64];         // stacked rows [sqrt(p_k) h_k ; y ; 0-pad]
    __shared__ float sWi[32 * 64];
    __shared__ float sDre[64 * 64];        // staging for uu tile, then D / Cholesky L
    __shared__ float sDim[64 * 64];
    __shared__ float sColR[64], sColI[64];
    __shared__ float sUr[16 * 64], sUi[16 * 64]; // forward-solve results
    __shared__ float sP[GG], sMask[GG];
    __shared__ float sGq[16], sNz[16];
    __shared__ float sRate[32];            // [0..15] UU rates, [16..31] DU rates
    __shared__ float sScal[4];             // alpha_lin, nBS, 1/ljj

    // ---------- Phase 0: async-stage the (16 x 64 complex) UUMatInit tile ----------
    // 8 KB contiguous DMA into the (currently dead) sDre region via the Tensor
    // Data Mover; it overlaps with the ~8K sincos VALU work in Phases A/B and
    // is consumed only at Phase C (after s_wait_tensorcnt + barrier).
    const float* uuTile = uu + (size_t)b * KK * NRR * 2;
#if defined(__has_builtin) && __has_builtin(__builtin_amdgcn_tensor_load_to_lds)
    if (tid < 32) {   // wave 0 issues the DMA (uniform descriptor in SGPRs)
        const unsigned int nElem  = KK * NRR * 2;                 // 2048 x 4B
        const unsigned int ldsOff = (unsigned int)(uintptr_t)(void*)sDre; // LDS aperture: low 32 bits = LDS byte offset
        const unsigned long long ga = (unsigned long long)(uintptr_t)(const void*)uuTile;
        u32x4 g0;
        g0.x = 1u;                                  // count=1, is_restore=0, gather off
        g0.y = ldsOff;                              // lds_addr
        g0.z = (unsigned int)ga;                    // global_addr[31:0]
        g0.w = (unsigned int)(ga >> 32) | (2u << 30); // global_addr[56:32] | type=2
        i32x8 g1;
        g1[0] = (int)(2u << 16);                    // workgroup_mask=0, data_size=4B
        g1[1] = (int)((nElem & 0xFFFFu) << 16);     // tensor_dim0[15:0]
        g1[2] = (int)((nElem >> 16) | (1u << 16));  // tensor_dim0[31:16] | tensor_dim1=1
        g1[3] = (int)((nElem & 0xFFFFu) << 16);     // tile_dim0 = nElem
        g1[4] = 0;                                  // tile_dim1/2 unused
        g1[5] = (int)nElem;                         // tensor_dim0_stride
        g1[6] = 0; g1[7] = 0;
        i32x4 g2 = {0, 0, 0, 0};
        i32x4 g3 = {0, 0, 0, 0};
#if __clang_major__ >= 23
        i32x8 g4 = g1; g4 = (i32x8)0;
        __builtin_amdgcn_tensor_load_to_lds(g0, g1, g2, g3, g4, 0);
#else
        __builtin_amdgcn_tensor_load_to_lds(g0, g1, g2, g3, 0);
#endif
    }
#else
    for (int e = tid; e < KK * NRR * 2; e += 256) sDre[e] = uuTile[e];
#endif

    // Hint the shared 64x64 phase table into cache early (global_prefetch_b8)
    if (tid < 64) __builtin_prefetch(rp + tid * NTT, 0, 0);

    // ---------- Phase A: x = sum_l DUCom + sum_m Sensing ; angles/gains ----------
    if (tid < NTT) {
        const int i = tid;
        float xr = 0.f, xi = 0.f;
        for (int l = 0; l < LLn; ++l) {
            const int e = ((b * LLn + l) * NTT + i) * 2;
            xr += duCom[e]; xi += duCom[e + 1];
        }
        for (int m = 0; m < MMn; ++m) {
            const int e = ((b * MMn + m) * NTT + i) * 2;
            xr += sens[e]; xi += sens[e + 1];
        }
        sX[2 * i] = xr; sX[2 * i + 1] = xi;
    }
    if (tid < 16) {
        float ang, g;
        if (tid < 8) {
            const int e = ((b * MMn + tid) * 2) * 2;       // real parts
            ang = ta[e]; g = ta[e + 2];
        } else {
            const int e = ((b * IIn + (tid - 8)) * 2) * 2;
            ang = inm[e]; g = inm[e + 2];
        }
        sAng[tid]  = ang;
        sGain[tid] = g;
        sPhi[tid]  = 3.14159265358979f * __sinf(ang * 0.017453292519943295f);
    }
    if (tid == 0) {
        sScal[0] = __powf(10.f, aSI[0] * 0.1f);   // alpha_lin
        sScal[1] = __powf(10.f, n2bs[0] * 0.1f);  // nBS
    }
    __syncthreads();

    // ---------- Phase B1: s_l = a_t(l)^H x ----------
    if (tid < 16) {
        const float phi = sPhi[tid];
        float sr = 0.f, si = 0.f;
        for (int t = 0; t < NTT; ++t) {
            float sn, cs; __sincosf(phi * (float)t, &sn, &cs);
            const float xr = sX[2 * t], xi = sX[2 * t + 1];
            sr += cs * xr + sn * xi;     // conj(a_t) * x
            si += cs * xi - sn * xr;
        }
        sSr[tid] = sr; sSi[tid] = si;
    }
    __syncthreads();

    // ---------- Phase B2: y = H_SI x + sum_l g_l a_r(l) s_l ----------
    if (tid < NRR) {
        const int r = tid;
        const float hscale = sqrtf(sScal[0]);
        float yr = 0.f, yi = 0.f;
        for (int t = 0; t < NTT; ++t) {
            const float ph = -6.283185307179586f * rp[r * NTT + t];
            float sn, cs; __sincosf(ph, &sn, &cs);
            const float xr = sX[2 * t], xi = sX[2 * t + 1];
            yr += cs * xr - sn * xi;
            yi += cs * xi + sn * xr;
        }
        yr *= hscale; yi *= hscale;
        for (int l = 0; l < 16; ++l) {
            float sn, cs; __sincosf(sPhi[l] * (float)r, &sn, &cs);
            const float g = sGain[l];
            yr += g * (cs * sSr[l] - sn * sSi[l]);
            yi += g * (cs * sSi[l] + sn * sSr[l]);
        }
        sY[2 * r] = yr; sY[2 * r + 1] = yi;
    }

    // Wait for the TDM transfer issued by wave 0, then make it block-visible.
#if defined(__has_builtin) && __has_builtin(__builtin_amdgcn_tensor_load_to_lds)
    if (tid < 32) __builtin_amdgcn_s_wait_tensorcnt(0);
#endif
    __syncthreads();

    // ---------- Phase C: build W (32x64) from staged tile, D = W^H W via WMMA ----------
    for (int e = tid; e < 32 * 64; e += 256) {
        const int row = e >> 6, c = e & 63;
        float wr = 0.f, wi = 0.f;
        if (row < 16) {
            const float sp = sqrtf(uuPow[b * KK + row]);
            const int g = (row * NRR + c) * 2;      // staged uu tile in sDre
            wr = sp * sDre[g]; wi = sp * sDre[g + 1];
        } else if (row == 16) {
            wr = sY[2 * c]; wi = sY[2 * c + 1];
        }
        sWr[e] = wr; sWi[e] = wi;
    }
    __syncthreads();

    {
        const int wv   = tid >> 5;
        const int lane = tid & 31;
        const int m    = lane & 15;
        const int half = lane >> 4;
        for (int rep = 0; rep < 2; ++rep) {
            const int tile = wv * 2 + rep;          // 16 tiles over 8 waves
            const int ti = tile >> 2, tj = tile & 3;
            const int rA = ti * 16 + m;             // W column for A fragment
            const int rB = tj * 16 + m;             // W column for B fragment
            v8f accRe = {}, accI1 = {}, accI2 = {};
            #pragma unroll
            for (int c = 0; c < 8; ++c) {           // inner dim 32 = 8 x K4
                const int kb = c * 4 + half * 2;
                v2f ar, ai, br, bi;
                ar.x = sWr[kb * 64 + rA]; ar.y = sWr[(kb + 1) * 64 + rA];
                ai.x = sWi[kb * 64 + rA]; ai.y = sWi[(kb + 1) * 64 + rA];
                br.x = sWr[kb * 64 + rB]; br.y = sWr[(kb + 1) * 64 + rB];
                bi.x = sWi[kb * 64 + rB]; bi.y = sWi[(kb + 1) * 64 + rB];
                // D_re = Wr^T Wr + Wi^T Wi ; D_im = Wr^T Wi - Wi^T Wr
                accRe = __builtin_amdgcn_wmma_f32_16x16x4_f32(false, ar, false, br, (short)0, accRe, false, false);
                accRe = __builtin_amdgcn_wmma_f32_16x16x4_f32(false, ai, false, bi, (short)0, accRe, false, false);
                accI1 = __builtin_amdgcn_wmma_f32_16x16x4_f32(false, ar, false, bi, (short)0, accI1, false, false);
                accI2 = __builtin_amdgcn_wmma_f32_16x16x4_f32(false, ai, false, br, (short)0, accI2, false, false);
            }
            #pragma unroll
            for (int v = 0; v < 8; ++v) {
                const int row = ti * 16 + v + 8 * half;
                const int col = tj * 16 + m;
                sDre[row * 64 + col] = accRe[v];
                sDim[row * 64 + col] = accI1[v] - accI2[v];
            }
        }
    }
    __syncthreads();
    if (tid < 64) sDre[tid * 64 + tid] += sScal[1];  // + nBS * I
    __syncthreads();

    // ---------- Phase D: in-place complex Cholesky of D (lower L) ----------
    for (int j = 0; j < 64; ++j) {
        if (tid == 0) {
            float d = sDre[j * 64 + j];
            d = d > 1e-20f ? d : 1e-20f;
            const float l = sqrtf(d);
            sDre[j * 64 + j] = l;
            sScal[2] = 1.0f / l;
        }
        __syncthreads();
        const float inv = sScal[2];
        if (tid > j && tid < 64) {
            const int i = tid;
            const float cr = sDre[i * 64 + j] * inv;
            const float cm = sDim[i * 64 + j] * inv;
            sDre[i * 64 + j] = cr; sDim[i * 64 + j] = cm;
            sColR[i] = cr; sColI[i] = cm;
        }
        __syncthreads();
        {
            const int i   = tid & 63;
            const int grp = tid >> 6;     // 4 column groups per row
            if (i > j) {
                const float cri = sColR[i], cii = sColI[i];
                for (int k = j + 1 + grp; k <= i; k += 4) {
                    const float crk = sColR[k], cik = sColI[k];
                    sDre[i * 64 + k] -= cri * crk + cii * cik;  // - col_i * conj(col_k)
                    sDim[i * 64 + k] -= cii * crk - cri * cik;
                }
            }
        }
        __syncthreads();
    }

    // ---------- Phase E/F/G in parallel across the block ----------
    if (tid < 16) {
        // UU: t_k = || L^{-1} h_k ||^2 ; rate = log2(1 + p_k t_k)  (Sherman-Morrison)
        const int k = tid;
        const float* h = uu + (size_t)((b * KK + k) * NRR) * 2;
        float t = 0.f;
        for (int r = 0; r < 64; ++r) {
            float sr_ = h[2 * r], si_ = h[2 * r + 1];
            for (int c = 0; c < r; ++c) {
                const float lr = sDre[r * 64 + c], li = sDim[r * 64 + c];
                const float ur = sUr[k * 64 + c], ui = sUi[k * 64 + c];
                sr_ -= lr * ur - li * ui;
                si_ -= lr * ui + li * ur;
            }
            const float inv = 1.0f / sDre[r * 64 + r];
            const float ur = sr_ * inv, ui = si_ * inv;
            sUr[k * 64 + r] = ur; sUi[k * 64 + r] = ui;
            t += ur * ur + ui * ui;
        }
        const float pk = uuPow[b * KK + k];
        sRate[k] = __log2f(1.0f + pk * t);
    } else if (tid < 32) {
        // DU: gHQg_l = |g_l^H x|^2 (rank-1 Q); interference terms
        const int l = tid - 16;
        float vr = 0.f, vi = 0.f;
        for (int i = 0; i < NTT; ++i) {
            const int e = ((b * LLn + l) * NTT + i) * 2;
            const float dr = du[e], di = du[e + 1];
            const float xr = sX[2 * i], xi = sX[2 * i + 1];
            vr += dr * xr + di * xi;
            vi += dr * xi - di * xr;
        }
        const float gq = vr * vr + vi * vi;
        float nz = 0.f;
        for (int k = 0; k < KK; ++k) {
            const int e = ((b * KK + k) * LLn + l) * 2;
            nz += uuPow[b * KK + k] * (ci[e] * ci[e] + ci[e + 1] * ci[e + 1]);
        }
        sGq[l] = gq;
        sNz[l] = nz + __powf(10.f, n2du[b * LLn + l] * 0.1f);
    } else {
        // Beampattern: P(g) = |a_g^H x|^2 over 181-point grid (rank-1 Q)
        const int g = tid - 32;
        if (g < GG) {
            const float phi = 3.14159265358979f * __sinf((float)g * 0.017453292519943295f);
            float ar = 0.f, ai = 0.f;
            for (int i = 0; i < NTT; ++i) {
                float sn, cs; __sincosf(phi * (float)i, &sn, &cs);
                const float xr = sX[2 * i], xi = sX[2 * i + 1];
                ar += cs * xr + sn * xi;
                ai += cs * xi - sn * xr;
            }
            sP[g] = ar * ar + ai * ai;
            float mk = 0.f;
            for (int m = 0; m < 8; ++m) {
                const float d = (float)g - sAng[m];
                if (d >= -10.0f && d <= 10.0f) mk = 1.0f;
            }
            sMask[g] = mk;
        }
    }
    __syncthreads();

    if (tid >= 16 && tid < 32) {
        const int l = tid - 16;
        float tot = 0.f;
        for (int j = 0; j < 16; ++j) tot += sGq[j];
        const float den = sNz[l] + tot - sGq[l];
        sRate[16 + l] = __log2f(1.0f + sGq[l] / den);
    }
    __syncthreads();

    // ---------- Final per-batch reduction & atomic mean ----------
    if (tid == 0) {
        float sr = 0.f;
        for (int k = 0; k < 32; ++k) sr += sRate[k];
        float bHb = 0.f, bHP = 0.f;
        for (int g = 0; g < GG; ++g) { bHb += sMask[g]; bHP += sMask[g] * sP[g]; }
        const float beta = bHP / (bHb + 1e-10f);
        float lo = 0.f;
        for (int g = 0; g < GG; ++g) {
            const float d = beta * sMask[g] - sP[g];
            lo += d * d;
        }
        lo *= (1.0f / (float)GG);
        const float contrib = (-sr + 100.0f * lo) * (1.0f / (float)BB);
        atomicAdd(out, contrib);
    }
}

extern "C" void kernel_launch(void* const* d_in, const int* in_sizes, int n_in,
                              void* d_out, int out_size, void* d_ws, size_t ws_size,
                              hipStream_t stream) {
    const float* uuPow = (const float*)d_in[0];
    const float* duCom = (const float*)d_in[1];
    const float* sens  = (const float*)d_in[2];
    const float* uu    = (const float*)d_in[3];
    const float* du    = (const float*)d_in[4];
    const float* ta    = (const float*)d_in[5];
    const float* inm   = (const float*)d_in[6];
    const float* ci    = (const float*)d_in[7];
    const float* n2du  = (const float*)d_in[8];
    const float* n2bs  = (const float*)d_in[9];
    const float* aSI   = (const float*)d_in[10];
    const float* rp    = (const float*)d_in[11];
    float* out = (float*)d_out;

    zero_out_kernel<<<1, 1, 0, stream>>>(out);
    loss_kernel<<<BB, 256, 0, stream>>>(uuPow, duCom, sens, uu, du, ta, inm, ci,
                                        n2du, n2bs, aSI, rp, out);
}